// Net_17514876633627
// MI455X (gfx1250) — compile-verified
//
#include <hip/hip_runtime.h>
#include <math.h>

#define N_NODES  10000
#define N_EDGES  160000
#define N_GRAPHS 64
#define D_IN     128
#define D_H      512
#define N_CLS    2
#define L_LAYERS 3
#define EPS_M    1e-7f
#define MAXV_M   10.0f
#define BN_EPS   1e-5f

typedef __attribute__((ext_vector_type(2))) float v2f;
typedef __attribute__((ext_vector_type(8))) float v8f;

// ---------------------------------------------------------------------------
// degree: deg[dst] += 1 per edge
// ---------------------------------------------------------------------------
__global__ void k_deg(const int* __restrict__ dst, float* __restrict__ deg, int nE) {
    int e = blockIdx.x * blockDim.x + threadIdx.x;
    if (e < nE) atomicAdd(&deg[dst[e]], 1.0f);
}

// ---------------------------------------------------------------------------
// edge messages: m = min(relu(x[src]) + eps, 10) ; agg[dst] += m^p
// one thread handles 4 consecutive features of one edge
// ---------------------------------------------------------------------------
__global__ void k_msg(const float* __restrict__ h, const int* __restrict__ src,
                      const int* __restrict__ dst, float* __restrict__ agg,
                      const float* __restrict__ pArr, int layer, int d, int nE) {
    int g4  = d >> 2;
    long long idx = (long long)blockIdx.x * blockDim.x + threadIdx.x;
    if (idx >= (long long)nE * g4) return;
    int e = (int)(idx / g4);
    int f = (int)(idx - (long long)e * g4) << 2;
    int s = src[e], t = dst[e];
    float p = pArr[layer];
    float4 xv = *reinterpret_cast<const float4*>(h + (size_t)s * d + f);
    float m0 = fminf(fmaxf(xv.x, 0.0f) + EPS_M, MAXV_M);
    float m1 = fminf(fmaxf(xv.y, 0.0f) + EPS_M, MAXV_M);
    float m2 = fminf(fmaxf(xv.z, 0.0f) + EPS_M, MAXV_M);
    float m3 = fminf(fmaxf(xv.w, 0.0f) + EPS_M, MAXV_M);
    if (p != 1.0f) { m0 = powf(m0, p); m1 = powf(m1, p); m2 = powf(m2, p); m3 = powf(m3, p); }
    float* ap = agg + (size_t)t * d + f;
    atomicAdd(ap + 0, m0);
    atomicAdd(ap + 1, m1);
    atomicAdd(ap + 2, m2);
    atomicAdd(ap + 3, m3);
}

// ---------------------------------------------------------------------------
// per-node: finish power-mean, MessageNorm, residual add -> hres
// one block (128 threads) per node
// ---------------------------------------------------------------------------
__global__ void k_node_update(const float* __restrict__ h, const float* __restrict__ agg,
                              const float* __restrict__ deg, const float* __restrict__ pArr,
                              const float* __restrict__ scArr, int layer, int d,
                              float* __restrict__ out) {
    __shared__ float sA[128];
    __shared__ float sH[128];
    const int node = blockIdx.x;
    const int tid  = threadIdx.x;
    const float p    = pArr[layer];
    const float sc   = scArr[layer];
    const float invp = 1.0f / p;
    const float invdeg = 1.0f / fmaxf(deg[node], 1.0f);
    const int nf = d >> 7;   // 1 (d=128) or 4 (d=512)

    float av[4], hv[4];
    float s_a = 0.0f, s_h = 0.0f;
    for (int j = 0; j < nf; ++j) {
        int f = tid + (j << 7);
        float a = agg[(size_t)node * d + f] * invdeg;
        a = fminf(fmaxf(a, EPS_M), MAXV_M);
        if (p != 1.0f) a = powf(a, invp);
        float x = h[(size_t)node * d + f];
        av[j] = a; hv[j] = x;
        s_a += a * a; s_h += x * x;
    }
    sA[tid] = s_a; sH[tid] = s_h; __syncthreads();
    for (int s = 64; s > 0; s >>= 1) {
        if (tid < s) { sA[tid] += sA[tid + s]; sH[tid] += sH[tid + s]; }
        __syncthreads();
    }
    const float anorm = sqrtf(sA[0]);
    const float xn    = sqrtf(sH[0]);
    const float inva  = 1.0f / fmaxf(anorm, 1e-12f);
    const float scale = inva * xn * sc;
    for (int j = 0; j < nf; ++j) {
        int f = tid + (j << 7);
        out[(size_t)node * d + f] = hv[j] + av[j] * scale;
    }
}

// ---------------------------------------------------------------------------
// pack W[K,N] into pair-interleaved Wp: Wp[((k>>1)*N + n)*2 + (k&1)] = W[k,n]
// so a lane's (K=kh, K=kh+1) B-fragment is one contiguous 8-byte load.
// ---------------------------------------------------------------------------
__global__ void k_pack_w(const float* __restrict__ W, float* __restrict__ Wp, int K, int N) {
    long long idx = (long long)blockIdx.x * blockDim.x + threadIdx.x;
    if (idx >= (long long)K * N) return;
    int k = (int)(idx / N);
    int n = (int)(idx - (long long)k * N);
    Wp[((size_t)(k >> 1) * N + n) * 2 + (k & 1)] = W[idx];
}

// ---------------------------------------------------------------------------
// C[M,N] = A[M,K] @ W[K,N] + bias[N] via V_WMMA_F32_16X16X4_F32
// W pre-packed by k_pack_w. blockDim = 128 (4 waves); each wave computes a
// 16-row x 128-col strip (8 accumulators -> 8 wmma per A fragment).
// Requires M % 16 == 0, K % 4 == 0, N % 128 == 0. EXEC uniform per wave.
// A 16x4 f32 layout: lanes 0-15 -> K={0,1}, lanes 16-31 -> K={2,3}
// ---------------------------------------------------------------------------
__global__ void k_gemm_wmma(const float* __restrict__ A, const float* __restrict__ Wp,
                            const float* __restrict__ bias, float* __restrict__ C,
                            int M, int K, int N) {
    const int lane = threadIdx.x & 31;
    const int wave = threadIdx.x >> 5;
    const int mt   = blockIdx.y * 4 + wave;        // 16-row tile index (wave-uniform)
    if (mt * 16 >= M) return;
    const int n0    = blockIdx.x * 128;
    const int m     = mt * 16 + (lane & 15);
    const int sel   = lane >> 4;                   // 0 or 1 (K-half select)
    const int khalf = sel << 1;                    // 0 or 2
    const int nl    = n0 + (lane & 15);

    const float* ap = A + (size_t)m * K + khalf;              // contiguous pair
    const float* bp = Wp + ((size_t)sel * N + nl) * 2;        // packed pair

    v8f acc[8] = {};
    for (int k = 0; k < K; k += 4) {
        v2f a = *reinterpret_cast<const v2f*>(ap);
#pragma unroll
        for (int j = 0; j < 8; ++j) {
            v2f b = *reinterpret_cast<const v2f*>(bp + j * 32);   // +16 cols = +32 floats
            acc[j] = __builtin_amdgcn_wmma_f32_16x16x4_f32(
                false, a, false, b, (short)0, acc[j], false, false);
        }
        ap += 4;
        bp += (size_t)4 * N;   // advance two packed pair-rows
    }

    // C layout: VGPR r -> row = r + 8*(lane/16), col = lane%16
    const int rbase = mt * 16 + (sel << 3);
    const int cb    = lane & 15;
#pragma unroll
    for (int j = 0; j < 8; ++j) {
        const int col = n0 + j * 16 + cb;
        const float bv = bias[col];
#pragma unroll
        for (int r = 0; r < 8; ++r) {
            C[(size_t)(rbase + r) * N + col] = acc[j][r] + bv;
        }
    }
}

// ---------------------------------------------------------------------------
// BatchNorm stats: per-feature mean / biased var over n rows (block per feature)
// ---------------------------------------------------------------------------
__global__ void k_bn_stats(const float* __restrict__ h, int n, int d,
                           float* __restrict__ mean, float* __restrict__ var) {
    __shared__ float sS[256];
    __shared__ float sQ[256];
    const int f = blockIdx.x;
    const int tid = threadIdx.x;
    float s = 0.0f, q = 0.0f;
    for (int r = tid; r < n; r += 256) {
        float v = h[(size_t)r * d + f];
        s += v; q += v * v;
    }
    sS[tid] = s; sQ[tid] = q; __syncthreads();
    for (int st = 128; st > 0; st >>= 1) {
        if (tid < st) { sS[tid] += sS[tid + st]; sQ[tid] += sQ[tid + st]; }
        __syncthreads();
    }
    if (tid == 0) {
        float mu = sS[0] / n;
        mean[f] = mu;
        var[f]  = sQ[0] / n - mu * mu;
    }
}

// ---------------------------------------------------------------------------
// BN apply + ReLU + JK-max (vectorized by 4 features)
// ---------------------------------------------------------------------------
__global__ void k_bn_apply(const float* __restrict__ hpre, const float* __restrict__ mean,
                           const float* __restrict__ var, const float* __restrict__ g,
                           const float* __restrict__ b, float* __restrict__ hout,
                           float* __restrict__ jk, int first, int n, int d) {
    int g4 = d >> 2;
    long long idx = (long long)blockIdx.x * blockDim.x + threadIdx.x;
    if (idx >= (long long)n * g4) return;
    int r = (int)(idx / g4);
    int f = (int)(idx - (long long)r * g4) << 2;
    size_t base = (size_t)r * d + f;
    float4 v = *reinterpret_cast<const float4*>(hpre + base);
    float4 o;
    o.x = fmaxf((v.x - mean[f + 0]) * rsqrtf(var[f + 0] + BN_EPS) * g[f + 0] + b[f + 0], 0.0f);
    o.y = fmaxf((v.y - mean[f + 1]) * rsqrtf(var[f + 1] + BN_EPS) * g[f + 1] + b[f + 1], 0.0f);
    o.z = fmaxf((v.z - mean[f + 2]) * rsqrtf(var[f + 2] + BN_EPS) * g[f + 2] + b[f + 2], 0.0f);
    o.w = fmaxf((v.w - mean[f + 3]) * rsqrtf(var[f + 3] + BN_EPS) * g[f + 3] + b[f + 3], 0.0f);
    *reinterpret_cast<float4*>(hout + base) = o;
    if (first) {
        *reinterpret_cast<float4*>(jk + base) = o;
    } else {
        float4 j = *reinterpret_cast<const float4*>(jk + base);
        j.x = fmaxf(j.x, o.x); j.y = fmaxf(j.y, o.y);
        j.z = fmaxf(j.z, o.z); j.w = fmaxf(j.w, o.w);
        *reinterpret_cast<float4*>(jk + base) = j;
    }
}

// ---------------------------------------------------------------------------
// pooling
// ---------------------------------------------------------------------------
__global__ void k_pool_count(const int* __restrict__ batch, float* __restrict__ cnt, int n) {
    int i = blockIdx.x * blockDim.x + threadIdx.x;
    if (i < n) atomicAdd(&cnt[batch[i]], 1.0f);
}

__global__ void k_pool_accum(const float* __restrict__ jk, const int* __restrict__ batch,
                             float* __restrict__ psum, unsigned int* __restrict__ pmax,
                             int n, int d) {
    int g4 = d >> 2;
    long long idx = (long long)blockIdx.x * blockDim.x + threadIdx.x;
    if (idx >= (long long)n * g4) return;
    int r = (int)(idx / g4);
    int f = (int)(idx - (long long)r * g4) << 2;
    int gph = batch[r];
    float4 v = *reinterpret_cast<const float4*>(jk + (size_t)r * d + f);
    size_t base = (size_t)gph * d + f;
    atomicAdd(&psum[base + 0], v.x);
    atomicAdd(&psum[base + 1], v.y);
    atomicAdd(&psum[base + 2], v.z);
    atomicAdd(&psum[base + 3], v.w);
    // jk >= 0 (post-ReLU) so uint compare == float compare
    atomicMax(&pmax[base + 0], __float_as_uint(v.x));
    atomicMax(&pmax[base + 1], __float_as_uint(v.y));
    atomicMax(&pmax[base + 2], __float_as_uint(v.z));
    atomicMax(&pmax[base + 3], __float_as_uint(v.w));
}

__global__ void k_pool_finalize(const float* __restrict__ psum, const unsigned int* __restrict__ pmax,
                                const float* __restrict__ cnt, float* __restrict__ gcat, int d) {
    int idx = blockIdx.x * blockDim.x + threadIdx.x;
    if (idx >= N_GRAPHS * d) return;
    int g = idx / d, f = idx - g * d;
    float c = cnt[g];
    float mx = (c > 0.0f) ? __uint_as_float(pmax[(size_t)g * d + f]) : 0.0f;
    float mn = psum[(size_t)g * d + f] / fmaxf(c, 1.0f);
    gcat[(size_t)g * 2 * d + f]     = mx;
    gcat[(size_t)g * 2 * d + d + f] = mn;
}

// ---------------------------------------------------------------------------
// BN over 64 rows (fc head) + ReLU: one block (64 threads) per feature
// ---------------------------------------------------------------------------
__global__ void k_bn_small(const float* __restrict__ in, const float* __restrict__ g,
                           const float* __restrict__ b, float* __restrict__ out,
                           int rows, int d) {
    __shared__ float sS[64];
    __shared__ float sQ[64];
    const int f = blockIdx.x;
    const int r = threadIdx.x;
    float v = in[(size_t)r * d + f];
    sS[r] = v; sQ[r] = v * v; __syncthreads();
    for (int st = 32; st > 0; st >>= 1) {
        if (r < st) { sS[r] += sS[r + st]; sQ[r] += sQ[r + st]; }
        __syncthreads();
    }
    float mu  = sS[0] / rows;
    float var = sQ[0] / rows - mu * mu;
    out[(size_t)r * d + f] = fmaxf((v - mu) * rsqrtf(var + BN_EPS) * g[f] + b[f], 0.0f);
}

// ---------------------------------------------------------------------------
// tiny fc2: [G,K] @ [K,C] + b
// ---------------------------------------------------------------------------
__global__ void k_fc2(const float* __restrict__ in, const float* __restrict__ W,
                      const float* __restrict__ b, float* __restrict__ out,
                      int G, int K, int C) {
    int t = blockIdx.x * blockDim.x + threadIdx.x;
    if (t >= G * C) return;
    int g = t / C, c = t - g * C;
    float acc = b[c];
    for (int k = 0; k < K; ++k) acc += in[(size_t)g * K + k] * W[(size_t)k * C + c];
    out[t] = acc;
}

// ---------------------------------------------------------------------------
static inline int cdiv(long long a, long long b) { return (int)((a + b - 1) / b); }

extern "C" void kernel_launch(void* const* d_in, const int* in_sizes, int n_in,
                              void* d_out, int out_size, void* d_ws, size_t ws_size,
                              hipStream_t stream) {
    const float* x     = (const float*)d_in[0];
    const float* W0    = (const float*)d_in[1];
    const float* Wrest = (const float*)d_in[2];
    const float* bconv = (const float*)d_in[3];
    const float* pArr  = (const float*)d_in[4];
    const float* msc   = (const float*)d_in[5];
    const float* bng   = (const float*)d_in[6];
    const float* bnb   = (const float*)d_in[7];
    const float* fc1W  = (const float*)d_in[8];
    const float* fc1b  = (const float*)d_in[9];
    const float* bn4g  = (const float*)d_in[10];
    const float* bn4b  = (const float*)d_in[11];
    const float* fc2W  = (const float*)d_in[12];
    const float* fc2b  = (const float*)d_in[13];
    const int*   ei    = (const int*)d_in[14];
    const int*   batch = (const int*)d_in[15];
    const int* src = ei;
    const int* dst = ei + N_EDGES;

    // ---- workspace carve-out (floats, 16B-aligned chunks) ----
    float* ws = (float*)d_ws;
    size_t off = 0;
    auto carve = [&](size_t n) { float* q = ws + off; off += (n + 3) & ~(size_t)3; return q; };
    float* deg   = carve(N_NODES);
    float* agg   = carve((size_t)N_NODES * D_H);
    float* hres  = carve((size_t)N_NODES * D_H);
    float* hbuf  = carve((size_t)N_NODES * D_H);
    float* hpre  = carve((size_t)N_NODES * D_H);
    float* jk    = carve((size_t)N_NODES * D_H);
    float* wpack = carve((size_t)2 * D_H * D_H);   // max K=1024 (fc1)
    float* mean  = carve(D_H);
    float* var   = carve(D_H);
    float* psum  = carve((size_t)N_GRAPHS * D_H);
    float* pmaxf = carve((size_t)N_GRAPHS * D_H);
    float* pcnt  = carve(N_GRAPHS);
    float* gcat  = carve((size_t)N_GRAPHS * 2 * D_H);
    float* fc1o  = carve((size_t)N_GRAPHS * D_H);
    float* bn4o  = carve((size_t)N_GRAPHS * D_H);
    (void)ws_size; (void)in_sizes; (void)n_in; (void)out_size;

    // degree (same for all layers)
    hipMemsetAsync(deg, 0, N_NODES * sizeof(float), stream);
    k_deg<<<cdiv(N_EDGES, 256), 256, 0, stream>>>(dst, deg, N_EDGES);

    const float* hcur = x;
    int d = D_IN;
    for (int l = 0; l < L_LAYERS; ++l) {
        hipMemsetAsync(agg, 0, (size_t)N_NODES * d * sizeof(float), stream);
        k_msg<<<cdiv((long long)N_EDGES * (d >> 2), 256), 256, 0, stream>>>(
            hcur, src, dst, agg, pArr, l, d, N_EDGES);
        k_node_update<<<N_NODES, 128, 0, stream>>>(hcur, agg, deg, pArr, msc, l, d, hres);

        const float* W = (l == 0) ? W0 : (Wrest + (size_t)(l - 1) * D_H * D_H);
        k_pack_w<<<cdiv((long long)d * D_H, 256), 256, 0, stream>>>(W, wpack, d, D_H);
        dim3 ggrid(D_H / 128, cdiv(N_NODES / 16, 4));
        k_gemm_wmma<<<ggrid, 128, 0, stream>>>(hres, wpack, bconv + (size_t)l * D_H, hpre,
                                               N_NODES, d, D_H);

        k_bn_stats<<<D_H, 256, 0, stream>>>(hpre, N_NODES, D_H, mean, var);
        k_bn_apply<<<cdiv((long long)N_NODES * (D_H >> 2), 256), 256, 0, stream>>>(
            hpre, mean, var, bng + (size_t)l * D_H, bnb + (size_t)l * D_H,
            hbuf, jk, (l == 0) ? 1 : 0, N_NODES, D_H);
        hcur = hbuf;
        d = D_H;
    }

    // ---- pooling ----
    hipMemsetAsync(psum,  0, (size_t)N_GRAPHS * D_H * sizeof(float), stream);
    hipMemsetAsync(pmaxf, 0, (size_t)N_GRAPHS * D_H * sizeof(float), stream);
    hipMemsetAsync(pcnt,  0, N_GRAPHS * sizeof(float), stream);
    k_pool_count<<<cdiv(N_NODES, 256), 256, 0, stream>>>(batch, pcnt, N_NODES);
    k_pool_accum<<<cdiv((long long)N_NODES * (D_H >> 2), 256), 256, 0, stream>>>(
        jk, batch, psum, (unsigned int*)pmaxf, N_NODES, D_H);
    k_pool_finalize<<<cdiv((long long)N_GRAPHS * D_H, 256), 256, 0, stream>>>(
        psum, (const unsigned int*)pmaxf, pcnt, gcat, D_H);

    // ---- head: fc1 (WMMA), BN+ReLU, fc2 ----
    k_pack_w<<<cdiv((long long)2 * D_H * D_H, 256), 256, 0, stream>>>(fc1W, wpack, 2 * D_H, D_H);
    {
        dim3 ggrid(D_H / 128, 1);   // M = 64 -> 4 row tiles = 4 waves of one block
        k_gemm_wmma<<<ggrid, 128, 0, stream>>>(gcat, wpack, fc1b, fc1o,
                                               N_GRAPHS, 2 * D_H, D_H);
    }
    k_bn_small<<<D_H, 64, 0, stream>>>(fc1o, bn4g, bn4b, bn4o, N_GRAPHS, D_H);
    k_fc2<<<1, 128, 0, stream>>>(bn4o, fc2W, fc2b, (float*)d_out,
                                 N_GRAPHS, D_H, N_CLS);
}